// InvariantPointAttention_52252572123767
// MI455X (gfx1250) — compile-verified
//
#include <hip/hip_runtime.h>
#include <math.h>

// ---------------------------------------------------------------------------
// Invariant Point Attention for MI455X (gfx1250).
// f32 WMMA (16x16x4) pipeline + Tensor Data Mover (TDM) async LDS staging.
// Problem is HBM-bound on z (134MB x 2 passes, ~11.5us floor @ 23.3TB/s);
// compute is ~2.5 GFLOP, so full-f32 matrix cores cost nothing vs bandwidth.
// B/A fragments (except ipa_bias's wb) come from LDS (ds_load_2addr pairs); softmax aside, GEMM tiles are
// filled by TDM descriptors (tensor_load_to_lds, 6-arg clang-23 form) with
// s_wait_tensorcnt, double-buffered in the two hot GEMMs.
// ---------------------------------------------------------------------------

#define NN  512
#define CS  384
#define CZ  128
#define CH  16
#define HHH 12

typedef float v2f __attribute__((ext_vector_type(2)));
typedef float v8f __attribute__((ext_vector_type(8)));
typedef unsigned int u32x4 __attribute__((ext_vector_type(4)));
typedef int i32x8 __attribute__((ext_vector_type(8)));
typedef int i32x4 __attribute__((ext_vector_type(4)));

#if defined(__has_builtin)
#if __has_builtin(__builtin_amdgcn_tensor_load_to_lds) && __has_builtin(__builtin_amdgcn_s_wait_tensorcnt)
#define USE_TDM 1
#endif
#endif
#ifndef USE_TDM
#define USE_TDM 0
#endif

// workspace layout (floats)
constexpr int OFF_Q    = 0;                       // 512*192
constexpr int OFF_K    = OFF_Q   + NN * 192;
constexpr int OFF_V    = OFF_K   + NN * 192;
constexpr int OFF_QP   = OFF_V   + NN * 192;      // 512*144 (local -> global in place)
constexpr int OFF_KP   = OFF_QP  + NN * 144;
constexpr int OFF_VPL  = OFF_KP  + NN * 144;      // 512*288 local vp
constexpr int OFF_VPG  = OFF_VPL + NN * 288;      // 512*384 padded global vp (32 cols/head)
constexpr int OFF_QN2  = OFF_VPG + NN * 384;      // 512*12
constexpr int OFF_KN2  = OFF_QN2 + NN * HHH;
constexpr int OFF_FEAT = OFF_KN2 + NN * HHH;      // 512*2112
constexpr int OFF_PRB  = OFF_FEAT + NN * 2112;    // 12 * 512 * 512 (bias, then probs)

static __device__ __forceinline__ v8f v8zero() {
  v8f z = {0.f,0.f,0.f,0.f,0.f,0.f,0.f,0.f}; return z;
}
static __device__ __forceinline__ v8f wmma4(v2f a, v2f b, v8f c) {
  // D = A(16x4 f32) x B(4x16 f32) + C(16x16 f32)
  return __builtin_amdgcn_wmma_f32_16x16x4_f32(false, a, false, b, (short)0, c, false, false);
}
// A fragment: rows at p, row-major stride lda, K cols [0..3] relative to p
static __device__ __forceinline__ v2f ldA(const float* p, int lda, int lane) {
  int m = lane & 15, kk = (lane >> 4) << 1;
  v2f a; a.x = p[m * lda + kk]; a.y = p[m * lda + kk + 1]; return a;
}
// B fragment: row-major B[k][n], stride ldb
static __device__ __forceinline__ v2f ldB(const float* p, int ldb, int lane) {
  int n = lane & 15, kk = (lane >> 4) << 1;
  v2f b; b.x = p[kk * ldb + n]; b.y = p[(kk + 1) * ldb + n]; return b;
}
// B fragment where logical B[k][n] = src[n][k] (src row-major, stride ld)
static __device__ __forceinline__ v2f ldBt(const float* p, int ld, int lane) {
  int n = lane & 15, kk = (lane >> 4) << 1;
  v2f b; b.x = p[n * ld + kk]; b.y = p[n * ld + kk + 1]; return b;
}

// ---------------------------------------------------------------------------
// TDM: async 2D f32 tile load global->LDS.
// Tile = rows x cols elements, source row pitch = stride elements.
// D# per CDNA5 ISA ch.8: tensor dims == tile dims (no OOB), data_size=4B.
// ---------------------------------------------------------------------------
#if USE_TDM
static __device__ __forceinline__ void tdm2d(float* lds, const float* g,
                                             int cols, int rows, int stride) {
  unsigned long long ga = (unsigned long long)(const void*)g;   // global byte addr
  unsigned int la = (unsigned int)(unsigned long long)(void*)lds; // LDS byte addr (low 32 of flat)
  u32x4 g0;
  g0[0] = 1u;                                                   // count=1, user D#
  g0[1] = la;                                                   // lds_addr
  g0[2] = (unsigned int)ga;                                     // global_addr[31:0]
  g0[3] = (unsigned int)((ga >> 32) & 0x1ffffffULL) | (2u << 30); // addr[56:32] | type=2
  i32x8 g1;
  g1[0] = 2 << 16;                                              // data_size=4B, no mask/pad
  g1[1] = (int)(((unsigned)cols & 0xffffu) << 16);              // tensor_dim0[15:0]
  g1[2] = (int)((((unsigned)cols >> 16) & 0xffffu) |
                (((unsigned)rows & 0xffffu) << 16));            // dim0 hi | tensor_dim1 lo
  g1[3] = (int)((((unsigned)rows >> 16) & 0xffffu) |
                (((unsigned)cols & 0xffffu) << 16));            // dim1 hi | tile_dim0
  g1[4] = (int)((unsigned)rows & 0xffffu);                      // tile_dim1, tile_dim2=0
  g1[5] = stride;                                               // tensor_dim0_stride[31:0]
  g1[6] = 0;                                                    // stride hi | dim1_stride lo
  g1[7] = 0;
  i32x4 gz4 = {0, 0, 0, 0};
  i32x8 gz8 = {0, 0, 0, 0, 0, 0, 0, 0};
  __builtin_amdgcn_tensor_load_to_lds(g0, g1, gz4, gz4, gz8, 0);
}
#define STAGE2D(lds, g, cols, rows, stride, lane) tdm2d((lds), (g), (cols), (rows), (stride))
#define TWAIT(n) __builtin_amdgcn_s_wait_tensorcnt((short)(n))
#else
static __device__ __forceinline__ void stage2d_sw(float* lds, const float* g,
                                                  int cols, int rows, int stride, int lane) {
  for (int i = lane; i < rows * cols; i += 32)
    lds[i] = g[(i / cols) * stride + (i % cols)];
}
#define STAGE2D(lds, g, cols, rows, stride, lane) stage2d_sw((lds), (g), (cols), (rows), (stride), (lane))
#define TWAIT(n) do { } while (0)
#endif

// ---------------------------------------------------------------------------
// Kernel 1: projections  s(512x384) @ {wq,wk,wv,wqp,wkp,wvp} + bias
// grid.x = 72 column tiles (12+12+12+9+9+18), grid.y = 32 row tiles, 1 wave.
// A (s rows) and B (W column panel) both TDM-staged into LDS per 128-K chunk.
// ---------------------------------------------------------------------------
__global__ void ipa_proj(const float* __restrict__ s,
    const float* wq, const float* bq, const float* wk, const float* bk,
    const float* wv, const float* bv, const float* wqp, const float* bqp,
    const float* wkp, const float* bkp, const float* wvp, const float* bvp,
    float* __restrict__ ws) {
  __shared__ float sa[16 * 128];
  __shared__ float wt[128 * 16];
  int lane = threadIdx.x;
  int ct = blockIdx.x, nb = blockIdx.y * 16;
  const float* W; const float* Bi; float* O; int ncol, cb;
  if (ct < 12)      { W = wq;  Bi = bq;  O = ws + OFF_Q;   ncol = 192; cb = ct * 16; }
  else if (ct < 24) { W = wk;  Bi = bk;  O = ws + OFF_K;   ncol = 192; cb = (ct - 12) * 16; }
  else if (ct < 36) { W = wv;  Bi = bv;  O = ws + OFF_V;   ncol = 192; cb = (ct - 24) * 16; }
  else if (ct < 45) { W = wqp; Bi = bqp; O = ws + OFF_QP;  ncol = 144; cb = (ct - 36) * 16; }
  else if (ct < 54) { W = wkp; Bi = bkp; O = ws + OFF_KP;  ncol = 144; cb = (ct - 45) * 16; }
  else              { W = wvp; Bi = bvp; O = ws + OFF_VPL; ncol = 288; cb = (ct - 54) * 16; }
  v8f acc = v8zero();
  for (int k0 = 0; k0 < CS; k0 += 128) {
    __syncthreads();
    STAGE2D(sa, s + nb * CS + k0, 128, 16, CS, lane);              // A: 16 x 128
    STAGE2D(wt, W + k0 * ncol + cb, 16, 128, ncol, lane);          // B: 128 x 16
    TWAIT(0);
    __syncthreads();
    for (int k = 0; k < 128; k += 4) {
      v2f a = ldA(&sa[k], 128, lane);
      v2f b = ldB(&wt[k * 16], 16, lane);
      acc = wmma4(a, b, acc);
    }
  }
  int c = lane & 15, hi = lane >> 4;
#pragma unroll
  for (int vg = 0; vg < 8; ++vg) {
    int r = vg + 8 * hi;
    O[(nb + r) * ncol + cb + c] = acc[vg] + Bi[cb + c];
  }
}

// ---------------------------------------------------------------------------
// Kernel 2: frame transform  g = R*l + t  (qp,kp in place; vp -> padded vpg)
// also per-(n,h) squared norms of qpg/kpg. 1 block per residue n, 64 threads.
// ---------------------------------------------------------------------------
__global__ void ipa_frames(const float* __restrict__ rot, const float* __restrict__ trn,
                           float* __restrict__ ws) {
  int n = blockIdx.x, tid = threadIdx.x;
  __shared__ float sq[144], sk[144];
  float R[9], t[3];
#pragma unroll
  for (int i = 0; i < 9; ++i) R[i] = rot[n * 9 + i];
#pragma unroll
  for (int i = 0; i < 3; ++i) t[i] = trn[n * 3 + i];
  float* qp = ws + OFF_QP + n * 144;
  float* kp = ws + OFF_KP + n * 144;
  const float* vpl = ws + OFF_VPL + n * 288;
  float* vpg = ws + OFF_VPG + n * 384;
  for (int task = tid; task < 192; task += 64) {
    float l0, l1, l2;
    if (task < 48)      { l0 = qp[task*3]; l1 = qp[task*3+1]; l2 = qp[task*3+2]; }
    else if (task < 96) { int u = task-48; l0 = kp[u*3]; l1 = kp[u*3+1]; l2 = kp[u*3+2]; }
    else                { int u = task-96; l0 = vpl[u*3]; l1 = vpl[u*3+1]; l2 = vpl[u*3+2]; }
    float g0 = R[0]*l0 + R[1]*l1 + R[2]*l2 + t[0];
    float g1 = R[3]*l0 + R[4]*l1 + R[5]*l2 + t[1];
    float g2 = R[6]*l0 + R[7]*l1 + R[8]*l2 + t[2];
    if (task < 48) {
      qp[task*3] = g0; qp[task*3+1] = g1; qp[task*3+2] = g2;
      sq[task*3] = g0; sq[task*3+1] = g1; sq[task*3+2] = g2;
    } else if (task < 96) {
      int u = task-48;
      kp[u*3] = g0; kp[u*3+1] = g1; kp[u*3+2] = g2;
      sk[u*3] = g0; sk[u*3+1] = g1; sk[u*3+2] = g2;
    } else {
      int u = task-96; int h = u >> 3, p = u & 7;
      vpg[h*32 + p*3 + 0] = g0; vpg[h*32 + p*3 + 1] = g1; vpg[h*32 + p*3 + 2] = g2;
    }
  }
  for (int i = tid; i < 96; i += 64) { int h = i >> 3; vpg[h*32 + 24 + (i & 7)] = 0.f; }
  __syncthreads();
  if (tid < 24) {
    int h = tid >> 1; bool isk = tid & 1;
    const float* sp = isk ? sk : sq;
    float s2 = 0.f;
#pragma unroll
    for (int j = 0; j < 12; ++j) { float x = sp[h*12 + j]; s2 += x * x; }
    ws[(isk ? OFF_KN2 : OFF_QN2) + n * HHH + h] = s2;
  }
}

// ---------------------------------------------------------------------------
// Kernel 3: pair bias  z(262144x128) @ wb(128x12) + bb  -> probs[h][n*512+m]
// grid = 16384 row tiles of 16, 1 wave. z tile TDM-staged (contiguous).
// ---------------------------------------------------------------------------
__global__ void ipa_bias(const float* __restrict__ z, const float* __restrict__ wb,
                         const float* __restrict__ bb, float* __restrict__ ws) {
  __shared__ float zt[16 * 128];
  int lane = threadIdx.x;
  int mb = blockIdx.x * 16;
  STAGE2D(zt, z + mb * CZ, 128, 16, CZ, lane);
  TWAIT(0);
  __syncthreads();
  v8f acc = v8zero();
  int n = lane & 15, hi = lane >> 4;
  for (int k = 0; k < CZ; k += 4) {
    v2f a = ldA(&zt[k], 128, lane);
    int kk = k + (hi << 1);
    v2f b;
    b.x = (n < HHH) ? wb[kk * HHH + n]       : 0.f;
    b.y = (n < HHH) ? wb[(kk + 1) * HHH + n] : 0.f;
    acc = wmma4(a, b, acc);
  }
  float* probs = ws + OFF_PRB;
  if (n < HHH) {
    float add = bb[n];
#pragma unroll
    for (int vg = 0; vg < 8; ++vg) {
      int nm = mb + vg + 8 * hi;
      probs[n * (NN * NN) + nm] = acc[vg] + add;
    }
  }
}

// ---------------------------------------------------------------------------
// Kernel 4: logits + softmax, in place over the bias plane.
// logits = 0.25*q.k + qpg.kpg - 0.5(|qpg|^2+|kpg|^2) + bias, mask, softmax(m).
// grid (32 query tiles, 12 heads), 1 wave.
// ---------------------------------------------------------------------------
__global__ void ipa_softmax(const int* __restrict__ mask, float* __restrict__ ws) {
  __shared__ float Lg[16][512];
  __shared__ float sq[16][16], sqp[16][12], skt[16][16], skp[16][12];
  __shared__ float sqn[16], skn[16];
  __shared__ int   smn[16], smm[16];
  int lane = threadIdx.x;
  int nb = blockIdx.x * 16, h = blockIdx.y;
  const float* q   = ws + OFF_Q;
  const float* k   = ws + OFF_K;
  const float* qpg = ws + OFF_QP;
  const float* kpg = ws + OFF_KP;
  const float* qn2 = ws + OFF_QN2;
  const float* kn2 = ws + OFF_KN2;
  float* ph = ws + OFF_PRB + h * (NN * NN);
  for (int i = lane; i < 256; i += 32) { int r = i >> 4, c = i & 15; sq[r][c] = q[(nb + r) * 192 + h * 16 + c]; }
  for (int i = lane; i < 192; i += 32) { int r = i / 12, c = i % 12; sqp[r][c] = qpg[(nb + r) * 144 + h * 12 + c]; }
  if (lane < 16) { sqn[lane] = qn2[(nb + lane) * HHH + h]; smn[lane] = mask[nb + lane]; }
  __syncthreads();
  int c = lane & 15, hi = lane >> 4;
  for (int mt = 0; mt < 32; ++mt) {
    int mb = mt * 16;
    __syncthreads();
    for (int i = lane; i < 256; i += 32) { int r = i >> 4, cc = i & 15; skt[r][cc] = k[(mb + r) * 192 + h * 16 + cc]; }
    for (int i = lane; i < 192; i += 32) { int r = i / 12, cc = i % 12; skp[r][cc] = kpg[(mb + r) * 144 + h * 12 + cc]; }
    if (lane < 16) { skn[lane] = kn2[(mb + lane) * HHH + h]; smm[lane] = mask[mb + lane]; }
    __syncthreads();
    v8f acc = v8zero();
#pragma unroll
    for (int ks = 0; ks < 16; ks += 4) {
      v2f a = ldA(&sq[0][ks], 16, lane);
      a.x *= 0.25f; a.y *= 0.25f;            // 1/sqrt(C_H)
      v2f b = ldBt(&skt[0][ks], 16, lane);
      acc = wmma4(a, b, acc);
    }
#pragma unroll
    for (int ks = 0; ks < 12; ks += 4) {
      v2f a = ldA(&sqp[0][ks], 12, lane);
      v2f b = ldBt(&skp[0][ks], 12, lane);
      acc = wmma4(a, b, acc);
    }
#pragma unroll
    for (int vg = 0; vg < 8; ++vg) {
      int r = vg + 8 * hi; int mc = mb + c;
      float lg = acc[vg] - 0.5f * (sqn[r] + skn[c]) + ph[(nb + r) * NN + mc];
      if (smn[r] == 0 || smm[c] == 0) lg = -1e30f;
      Lg[r][mc] = lg;
    }
  }
  __syncthreads();
  for (int r = 0; r < 16; ++r) {
    float mx = -1e30f;
    for (int m = lane; m < NN; m += 32) mx = fmaxf(mx, Lg[r][m]);
    for (int o = 16; o; o >>= 1) mx = fmaxf(mx, __shfl_xor(mx, o, 32));
    float sum = 0.f;
    for (int m = lane; m < NN; m += 32) { float e = __expf(Lg[r][m] - mx); Lg[r][m] = e; sum += e; }
    for (int o = 16; o; o >>= 1) sum += __shfl_xor(sum, o, 32);
    float inv = 1.f / sum;
    float* dst = ph + (nb + r) * NN;
    for (int m = lane; m < NN; m += 32) dst[m] = Lg[r][m] * inv;
  }
}

// ---------------------------------------------------------------------------
// Kernel 5: per (n-tile, h): w @ v (16 cols) and w @ vpg (32 padded cols),
// K = 512 via WMMA. probs tile + v/vpg column panels all TDM-staged.
// Epilogue applies R^T(x - t) and point norms.
// ---------------------------------------------------------------------------
__global__ void ipa_attn_v(const float* __restrict__ rot, const float* __restrict__ trn,
                           float* __restrict__ ws) {
  __shared__ float sp[16 * 512];
  __shared__ float lv[512 * 16];
  __shared__ float lvp[512 * 32];
  __shared__ float pts[16][32];
  int lane = threadIdx.x;
  int nb = blockIdx.x * 16, h = blockIdx.y;
  const float* ph  = ws + OFF_PRB + h * (NN * NN);
  const float* v   = ws + OFF_V;
  const float* vpg = ws + OFF_VPG;
  float* feat = ws + OFF_FEAT;
  STAGE2D(sp,  ph + nb * NN, 512, 16, NN, lane);    // A: probs 16 x 512
  STAGE2D(lv,  v + h * 16,   16, 512, 192, lane);   // B: v panel 512 x 16
  STAGE2D(lvp, vpg + h * 32, 32, 512, 384, lane);   // B: vpg panel 512 x 32
  TWAIT(0);
  __syncthreads();
  v8f aS = v8zero(), aP0 = v8zero(), aP1 = v8zero();
  for (int k0 = 0; k0 < NN; k0 += 4) {
    v2f a  = ldA(&sp[k0], 512, lane);
    v2f b0 = ldB(&lv[k0 * 16], 16, lane);
    v2f b1 = ldB(&lvp[k0 * 32], 32, lane);
    v2f b2 = ldB(&lvp[k0 * 32 + 16], 32, lane);
    aS  = wmma4(a, b0, aS);
    aP0 = wmma4(a, b1, aP0);
    aP1 = wmma4(a, b2, aP1);
  }
  int c = lane & 15, hi = lane >> 4;
#pragma unroll
  for (int vg = 0; vg < 8; ++vg) {
    int r = vg + 8 * hi; int n = nb + r;
    feat[n * 2112 + h * 176 + c] = aS[vg];   // v_out_scalar
    pts[r][c]      = aP0[vg];
    pts[r][16 + c] = aP1[vg];
  }
  __syncthreads();
  for (int tsk = lane; tsk < 128; tsk += 32) {   // 16 queries x 8 points
    int qr = tsk >> 3, p = tsk & 7;
    int n = nb + qr;
    float g0 = pts[qr][p*3 + 0] - trn[n*3 + 0];
    float g1 = pts[qr][p*3 + 1] - trn[n*3 + 1];
    float g2 = pts[qr][p*3 + 2] - trn[n*3 + 2];
    const float* R = rot + n * 9;
    float l0 = R[0]*g0 + R[3]*g1 + R[6]*g2;    // R^T
    float l1 = R[1]*g0 + R[4]*g1 + R[7]*g2;
    float l2 = R[2]*g0 + R[5]*g1 + R[8]*g2;
    float* fb = feat + n * 2112 + h * 176;
    fb[16 + p*3 + 0] = l0; fb[16 + p*3 + 1] = l1; fb[16 + p*3 + 2] = l2;
    fb[40 + p] = sqrtf(l0*l0 + l1*l1 + l2*l2);
  }
}

// ---------------------------------------------------------------------------
// Kernel 6: pair features. Per query n: (12x512) probs^T @ z[n] (512x128).
// probs columns TDM-staged once; z streamed in double-buffered TDM chunks
// (16 rows x 128) overlapped with WMMA via s_wait_tensorcnt 1.
// ---------------------------------------------------------------------------
static __device__ __forceinline__ void pair_consume(const float* sp, const float* zl,
                                                    v8f* acc, int c, int lane) {
  int hr = lane & 15, hi = lane >> 4;
  for (int k = 0; k < 16; k += 4) {
    int kk = c * 16 + k + (hi << 1);
    v2f a;
    a.x = (hr < HHH) ? sp[hr * NN + kk]     : 0.f;
    a.y = (hr < HHH) ? sp[hr * NN + kk + 1] : 0.f;
#pragma unroll
    for (int t = 0; t < 8; ++t) {
      v2f b = ldB(zl + k * CZ + t * 16, CZ, lane);
      acc[t] = wmma4(a, b, acc[t]);
    }
  }
}

__global__ void ipa_pair(const float* __restrict__ z, float* __restrict__ ws) {
  __shared__ float sp[HHH * NN];          // 12 x 512 probs columns for this query
  __shared__ float zb[2][16 * CZ];        // double-buffered z chunks
  int lane = threadIdx.x;
  int n = blockIdx.x;
  const float* probs = ws + OFF_PRB;
  float* feat = ws + OFF_FEAT;
  const float* zr = z + (long)n * NN * CZ;
  STAGE2D(sp, probs + n * NN, 512, HHH, NN * NN, lane);  // row h pitch = plane stride
  STAGE2D(zb[0], zr, CZ, 16, CZ, lane);
  v8f acc[8];
#pragma unroll
  for (int t = 0; t < 8; ++t) acc[t] = v8zero();
  for (int c = 0; c < 31; ++c) {
    STAGE2D(zb[(c + 1) & 1], zr + (c + 1) * 16 * CZ, CZ, 16, CZ, lane);
    TWAIT(1);                               // sp + chunk c complete; c+1 in flight
#if !USE_TDM
    __syncthreads();
#endif
    pair_consume(sp, zb[c & 1], acc, c, lane);
  }
  TWAIT(0);
#if !USE_TDM
  __syncthreads();
#endif
  pair_consume(sp, zb[1], acc, 31, lane);
  int hi = lane >> 4;
#pragma unroll
  for (int t = 0; t < 8; ++t) {
#pragma unroll
    for (int vg = 0; vg < 8; ++vg) {
      int hrow = vg + 8 * hi;
      if (hrow < HHH)
        feat[n * 2112 + hrow * 176 + 48 + t * 16 + (lane & 15)] = acc[t][vg];
    }
  }
}

// ---------------------------------------------------------------------------
// Kernel 7: output GEMM  features(512x2112) @ wo(2112x384) + bo
// grid (24 col tiles, 32 row tiles), 1 wave. A and B chunks (64-K) both
// TDM-staged, double-buffered, overlapped via s_wait_tensorcnt 2.
// ---------------------------------------------------------------------------
static __device__ __forceinline__ void out_consume(const float* sa, const float* sb,
                                                   v8f* acc, int lane) {
  for (int k = 0; k < 64; k += 4) {
    v2f a = ldA(&sa[k], 64, lane);
    v2f b = ldB(&sb[k * 16], 16, lane);
    *acc = wmma4(a, b, *acc);
  }
}

__global__ void ipa_out(const float* __restrict__ wo, const float* __restrict__ bo,
                        const float* __restrict__ ws, float* __restrict__ out) {
  __shared__ float sf[2][16 * 64];
  __shared__ float sw[2][64 * 16];
  int lane = threadIdx.x;
  int cb = blockIdx.x * 16, nb = blockIdx.y * 16;
  const float* feat = ws + OFF_FEAT;
  v8f acc = v8zero();
  STAGE2D(sf[0], feat + nb * 2112, 64, 16, 2112, lane);
  STAGE2D(sw[0], wo + cb, 16, 64, CS, lane);
  for (int c = 0; c < 32; ++c) {          // 33 chunks total (2112 = 33*64)
    STAGE2D(sf[(c + 1) & 1], feat + nb * 2112 + (c + 1) * 64, 64, 16, 2112, lane);
    STAGE2D(sw[(c + 1) & 1], wo + (c + 1) * 64 * CS + cb, 16, 64, CS, lane);
    TWAIT(2);                              // current pair done; next pair in flight
#if !USE_TDM
    __syncthreads();
#endif
    out_consume(sf[c & 1], sw[c & 1], &acc, lane);
  }
  TWAIT(0);
#if !USE_TDM
  __syncthreads();
#endif
  out_consume(sf[0], sw[0], &acc, lane);   // chunk 32 sits in buffer 32&1==0
  int c = lane & 15, hi = lane >> 4;
#pragma unroll
  for (int vg = 0; vg < 8; ++vg) {
    int r = vg + 8 * hi;
    out[(nb + r) * CS + cb + c] = acc[vg] + bo[cb + c];
  }
}

// ---------------------------------------------------------------------------
extern "C" void kernel_launch(void* const* d_in, const int* in_sizes, int n_in,
                              void* d_out, int out_size, void* d_ws, size_t ws_size,
                              hipStream_t stream) {
  const float* s    = (const float*)d_in[0];
  const float* z    = (const float*)d_in[1];
  const float* trn  = (const float*)d_in[2];
  const float* rot  = (const float*)d_in[3];
  const int*   mask = (const int*)  d_in[4];
  const float* wq   = (const float*)d_in[5];  const float* bq  = (const float*)d_in[6];
  const float* wk   = (const float*)d_in[7];  const float* bk  = (const float*)d_in[8];
  const float* wv   = (const float*)d_in[9];  const float* bv  = (const float*)d_in[10];
  const float* wqp  = (const float*)d_in[11]; const float* bqp = (const float*)d_in[12];
  const float* wkp  = (const float*)d_in[13]; const float* bkp = (const float*)d_in[14];
  const float* wvp  = (const float*)d_in[15]; const float* bvp = (const float*)d_in[16];
  const float* wb   = (const float*)d_in[17]; const float* bb  = (const float*)d_in[18];
  const float* wo   = (const float*)d_in[19]; const float* bo  = (const float*)d_in[20];
  float* ws  = (float*)d_ws;
  float* out = (float*)d_out;
  (void)in_sizes; (void)n_in; (void)out_size; (void)ws_size;

  ipa_proj   <<<dim3(72, 32), 32, 0, stream>>>(s, wq, bq, wk, bk, wv, bv,
                                               wqp, bqp, wkp, bkp, wvp, bvp, ws);
  ipa_frames <<<dim3(NN),     64, 0, stream>>>(rot, trn, ws);
  ipa_bias   <<<dim3(16384),  32, 0, stream>>>(z, wb, bb, ws);
  ipa_softmax<<<dim3(32, 12), 32, 0, stream>>>(mask, ws);
  ipa_attn_v <<<dim3(32, 12), 32, 0, stream>>>(rot, trn, ws);
  ipa_pair   <<<dim3(NN),     32, 0, stream>>>(z, ws);
  ipa_out    <<<dim3(24, 32), 32, 0, stream>>>(wo, bo, ws, out);
}